// SageModel_55241869361537
// MI455X (gfx1250) — compile-verified
//
#include <hip/hip_runtime.h>
#include <math.h>

typedef __attribute__((ext_vector_type(2))) float v2f;
typedef __attribute__((ext_vector_type(8))) float v8f;

#define IN_DIM 512

__device__ __forceinline__ v8f wmma_f32_4(v2f a, v2f b, v8f c) {
  // V_WMMA_F32_16X16X4_F32: D = A(16x4) * B(4x16) + C(16x16), fp32 exact
  return __builtin_amdgcn_wmma_f32_16x16x4_f32(false, a, false, b, (short)0, c,
                                               false, false);
}

// ---------------- zero workspace ----------------
__global__ void k_zero(float* p, int n) {
  int i = blockIdx.x * blockDim.x + threadIdx.x;
  if (i < n) p[i] = 0.0f;
}

// ---------------- fc1: h1 = relu(x @ W1 + b1), W1: [512,16] ----------------
__global__ __launch_bounds__(128) void k_fc1(const float* __restrict__ x,
                                             const float* __restrict__ w1,
                                             const float* __restrict__ b1,
                                             float* __restrict__ h1,
                                             int ntiles) {
  __shared__ float sW[IN_DIM * 16];  // 32 KB
  int tid = threadIdx.x;
  for (int i = tid; i < IN_DIM * 16; i += 128) sW[i] = w1[i];
  __syncthreads();

  int wave = tid >> 5, lane = tid & 31;
  int tile = blockIdx.x * 4 + wave;
  if (tile >= ntiles) return;  // wave-uniform: EXEC stays all-ones

  int n = lane & 15;               // A row (M) for this lane, and B/D column
  int koff = (lane >> 4) << 1;     // 0 or 2: K sub-offset per ISA layout
  long long row = (long long)tile * 16 + n;
  const float* xr = x + row * IN_DIM;

  v8f acc = {};
#pragma unroll 8
  for (int kb = 0; kb < IN_DIM / 4; ++kb) {
    int k = kb * 4 + koff;
    v2f a; a.x = xr[k]; a.y = xr[k + 1];
    v2f b; b.x = sW[k * 16 + n]; b.y = sW[(k + 1) * 16 + n];
    acc = wmma_f32_4(a, b, acc);
  }

  float bias = b1[n];
  int rbase = tile * 16 + ((lane >> 4) << 3);
#pragma unroll
  for (int r = 0; r < 8; ++r)
    h1[(long long)(rbase + r) * 16 + n] = fmaxf(acc[r] + bias, 0.0f);
}

// ---------------- edge scatter layer 1 (16 ch) + degree count ----------------
__global__ void k_scatter1(const int* __restrict__ src, const int* __restrict__ dst,
                           const float* __restrict__ h1, float* __restrict__ agg1,
                           float* __restrict__ cnt, int nE) {
  int t = blockIdx.x * blockDim.x + threadIdx.x;
  int e = t >> 4, c = t & 15;
  if (e >= nE) return;
  int s = src[e], d = dst[e];
  atomicAdd(&agg1[(long long)d * 16 + c], h1[(long long)s * 16 + c]);
  if (c == 0) atomicAdd(&cnt[d], 1.0f);
}

// ---------------- conv1: out1 = relu([mean|h1] @ [ll;lr] + llb), K=32 ----------
__global__ __launch_bounds__(128) void k_conv1(const float* __restrict__ agg1,
                                               const float* __restrict__ cnt,
                                               const float* __restrict__ h1,
                                               const float* __restrict__ llw,
                                               const float* __restrict__ llb,
                                               const float* __restrict__ lrw,
                                               float* __restrict__ out1,
                                               int ntiles) {
  __shared__ float sW[32 * 32];  // [K=32][Ncols=32]
  int tid = threadIdx.x;
  for (int i = tid; i < 32 * 32; i += 128) {
    int k = i >> 5, j = i & 31;
    sW[i] = (k < 16) ? llw[k * 32 + j] : lrw[(k - 16) * 32 + j];
  }
  __syncthreads();

  int wave = tid >> 5, lane = tid & 31;
  int tile = blockIdx.x * 4 + wave;
  if (tile >= ntiles) return;

  int n = lane & 15;
  int koff = (lane >> 4) << 1;
  long long row = (long long)tile * 16 + n;
  float inv = 1.0f / fmaxf(cnt[row], 1.0f);

  v8f c0 = {}, c1 = {};
#pragma unroll
  for (int kb = 0; kb < 8; ++kb) {
    int kk = kb * 4 + koff;  // boundary 16 aligned to step 4 -> wave-uniform side
    v2f a;
    if (kk < 16) { a.x = agg1[row * 16 + kk] * inv; a.y = agg1[row * 16 + kk + 1] * inv; }
    else         { a.x = h1[row * 16 + kk - 16];    a.y = h1[row * 16 + kk - 15]; }
    v2f b0; b0.x = sW[kk * 32 + n];      b0.y = sW[(kk + 1) * 32 + n];
    v2f b1; b1.x = sW[kk * 32 + 16 + n]; b1.y = sW[(kk + 1) * 32 + 16 + n];
    c0 = wmma_f32_4(a, b0, c0);
    c1 = wmma_f32_4(a, b1, c1);
  }

  float bl = llb[n], bh = llb[n + 16];
  int rbase = tile * 16 + ((lane >> 4) << 3);
#pragma unroll
  for (int r = 0; r < 8; ++r) {
    long long orow = (long long)(rbase + r) * 32;
    out1[orow + n]      = fmaxf(c0[r] + bl, 0.0f);
    out1[orow + 16 + n] = fmaxf(c1[r] + bh, 0.0f);
  }
}

// ---------------- edge scatter layer 2 (32 ch) ----------------
__global__ void k_scatter2(const int* __restrict__ src, const int* __restrict__ dst,
                           const float* __restrict__ out1, float* __restrict__ agg2,
                           int nE) {
  int t = blockIdx.x * blockDim.x + threadIdx.x;
  int e = t >> 5, c = t & 31;
  if (e >= nE) return;
  int s = src[e], d = dst[e];
  atomicAdd(&agg2[(long long)d * 32 + c], out1[(long long)s * 32 + c]);
}

// ---------------- conv2: out2 = relu([mean|out1] @ [ll;lr] + llb), K=64 --------
__global__ __launch_bounds__(128) void k_conv2(const float* __restrict__ agg2,
                                               const float* __restrict__ cnt,
                                               const float* __restrict__ out1,
                                               const float* __restrict__ llw,
                                               const float* __restrict__ llb,
                                               const float* __restrict__ lrw,
                                               float* __restrict__ out2,
                                               int ntiles) {
  __shared__ float sW[64 * 16];  // [K=64][Ncols=16]
  int tid = threadIdx.x;
  for (int i = tid; i < 64 * 16; i += 128) {
    int k = i >> 4, j = i & 15;
    sW[i] = (k < 32) ? llw[k * 16 + j] : lrw[(k - 32) * 16 + j];
  }
  __syncthreads();

  int wave = tid >> 5, lane = tid & 31;
  int tile = blockIdx.x * 4 + wave;
  if (tile >= ntiles) return;

  int n = lane & 15;
  int koff = (lane >> 4) << 1;
  long long row = (long long)tile * 16 + n;
  float inv = 1.0f / fmaxf(cnt[row], 1.0f);

  v8f acc = {};
#pragma unroll
  for (int kb = 0; kb < 16; ++kb) {
    int kk = kb * 4 + koff;  // boundary 32 aligned to step 4 -> wave-uniform side
    v2f a;
    if (kk < 32) { a.x = agg2[row * 32 + kk] * inv; a.y = agg2[row * 32 + kk + 1] * inv; }
    else         { a.x = out1[row * 32 + kk - 32]; a.y = out1[row * 32 + kk - 31]; }
    v2f b; b.x = sW[kk * 16 + n]; b.y = sW[(kk + 1) * 16 + n];
    acc = wmma_f32_4(a, b, acc);
  }

  float bias = llb[n];
  int rbase = tile * 16 + ((lane >> 4) << 3);
#pragma unroll
  for (int r = 0; r < 8; ++r)
    out2[(long long)(rbase + r) * 16 + n] = fmaxf(acc[r] + bias, 0.0f);
}

// ---------------- head: fc2 + sigmoid/relu combos ----------------
__global__ void k_head(const float* __restrict__ out2, const float* __restrict__ w2,
                       const float* __restrict__ b2, float* __restrict__ out, int N) {
  int i = blockIdx.x * blockDim.x + threadIdx.x;
  if (i >= N) return;
  float h0 = b2[0], h1 = b2[1], h2 = b2[2];
  const float* r = out2 + (long long)i * 16;
#pragma unroll
  for (int k = 0; k < 16; ++k) {
    float v = r[k];
    h0 += v * w2[k * 3 + 0];
    h1 += v * w2[k * 3 + 1];
    h2 += v * w2[k * 3 + 2];
  }
  float gsi = 1.0f / (1.0f + __expf(-h1));
  float mxi = 1.0f / (1.0f + __expf(-h2));
  float fsi = fmaxf(h0, 0.0f) + gsi;
  out[(long long)i * 3 + 0] = fsi;
  out[(long long)i * 3 + 1] = gsi;
  out[(long long)i * 3 + 2] = mxi;
}

extern "C" void kernel_launch(void* const* d_in, const int* in_sizes, int n_in,
                              void* d_out, int out_size, void* d_ws, size_t ws_size,
                              hipStream_t stream) {
  const float* x    = (const float*)d_in[0];
  const int*   ei   = (const int*)d_in[1];   // [2, E] int32
  const float* w1   = (const float*)d_in[2];
  const float* b1   = (const float*)d_in[3];
  const float* c1ll = (const float*)d_in[4];
  const float* c1lb = (const float*)d_in[5];
  const float* c1lr = (const float*)d_in[6];
  const float* c2ll = (const float*)d_in[7];
  const float* c2lb = (const float*)d_in[8];
  const float* c2lr = (const float*)d_in[9];
  const float* w2   = (const float*)d_in[10];
  const float* b2   = (const float*)d_in[11];
  float* out = (float*)d_out;

  int N = in_sizes[0] / IN_DIM;   // 100000 (divisible by 16)
  int E = in_sizes[1] / 2;        // 3200000
  const int* src = ei;
  const int* dst = ei + E;

  // Workspace layout (floats)
  float* ws   = (float*)d_ws;
  float* h1   = ws;                           // 16N
  float* cnt  = ws + (long long)16 * N;       // N
  float* agg1 = ws + (long long)17 * N;       // 16N
  float* out1 = ws + (long long)33 * N;       // 32N
  float* agg2 = ws + (long long)65 * N;       // 32N
  float* out2 = ws + (long long)97 * N;       // 16N

  int ntiles = (N + 15) / 16;
  int wblocks = (ntiles + 3) / 4;

  // zero cnt+agg1 (17N) and agg2 (32N) every call (determinism)
  {
    int n1 = 17 * N;
    k_zero<<<(n1 + 255) / 256, 256, 0, stream>>>(cnt, n1);
    int n2 = 32 * N;
    k_zero<<<(n2 + 255) / 256, 256, 0, stream>>>(agg2, n2);
  }

  k_fc1<<<wblocks, 128, 0, stream>>>(x, w1, b1, h1, ntiles);

  {
    long long t = (long long)E * 16;
    k_scatter1<<<(unsigned)((t + 255) / 256), 256, 0, stream>>>(src, dst, h1, agg1, cnt, E);
  }

  k_conv1<<<wblocks, 128, 0, stream>>>(agg1, cnt, h1, c1ll, c1lb, c1lr, out1, ntiles);

  {
    long long t = (long long)E * 32;
    k_scatter2<<<(unsigned)((t + 255) / 256), 256, 0, stream>>>(src, dst, out1, agg2, E);
  }

  k_conv2<<<wblocks, 128, 0, stream>>>(agg2, cnt, out1, c2ll, c2lb, c2lr, out2, ntiles);

  k_head<<<(N + 255) / 256, 256, 0, stream>>>(out2, w2, b2, out, N);
}